// MultiLevelLatticeProcessor_90606630076689
// MI455X (gfx1250) — compile-verified
//
#include <hip/hip_runtime.h>
#include <hip/hip_bf16.h>
#include <math.h>

// ---------------------------------------------------------------------------
// MultiLevelLatticeProcessor for MI455X (gfx1250).
// Static lattice (S=8192) baked into constant tables. All dense GEMMs run on
// V_WMMA_F32_16X16X4_F32 (fp32 in / fp32 acc) — the problem is bandwidth
// bound (dominated by the 128MB x->out copy), so fp32 WMMA keeps reference
// precision at zero cost.
//
// GEMM inner loop is unguarded: A rows >= M are loaded anyway (row m of A
// only affects row m of D, which the store guard discards), and every padded
// tile overread stays inside the workspace allocation by construction.
// ---------------------------------------------------------------------------

#define Bq   4
#define Sq   8192
#define Dm   1024
#define NPOS 7
#define NENT 16   // (pos,level) entries, sorted by level: 7 L0, 6 L1, 3 L2

typedef float v2f __attribute__((ext_vector_type(2)));
typedef float v8f __attribute__((ext_vector_type(8)));

// Entries sorted by level. Level row ranges in the 64-row (entry*4+b) space:
//   L0: rows [0,28)   L1: rows [28,52)   L2: rows [52,64)
__constant__ int   c_pos[NPOS]     = {12, 36, 104, 304, 888, 2592, 7568};
__constant__ int   c_nlev[NPOS]    = {1, 2, 2, 2, 3, 3, 3};
// kv entry index (into the level-sorted entry list) per (pos, level)
__constant__ int   c_kvidx[NPOS][3] = {
    {0, 0, 0}, {1, 7, 7}, {2, 8, 8}, {3, 9, 9},
    {4, 10, 13}, {5, 11, 14}, {6, 12, 15}};
__constant__ int   c_nn[NENT] = {3,3,3,3,3,3,3, 1,2,3,3,3,3, 1,2,3};
__constant__ int   c_nodes[NENT][3] = {
    {0,2,4},{2,4,12},{4,12,36},{12,36,104},{36,104,304},{104,304,888},{304,888,2592},
    {0,0,0},{0,2,0},{0,2,4},{2,4,12},{4,12,36},{12,36,104},
    {0,0,0},{0,2,0},{0,2,4}};
__constant__ float c_w[NENT][3] = {
    {1,1,1},{1,1,1},{1,1,1},{1,1,1},{1,1,1},{1,1,1},{1,1,1},
    {1,0,0},{1,1,0},{1,1,1},{1,2,3},{1,2,3},{1,2,3},
    {1,0,0},{1,1,0},{1,1,1}};
__constant__ float c_winv[NENT] = {
    1.f/3,1.f/3,1.f/3,1.f/3,1.f/3,1.f/3,1.f/3,
    1.f,  1.f/2,1.f/3,1.f/6,1.f/6,1.f/6,
    1.f,  1.f/2,1.f/3};

// ---------------------------------------------------------------------------
// out = x  (128 MB, the roofline-dominant term: ~11us at 23.3 TB/s)
// ---------------------------------------------------------------------------
__global__ void copy_kernel(const float4* __restrict__ in, float4* __restrict__ out, long n) {
  long i = (long)blockIdx.x * blockDim.x + threadIdx.x;
  long stride = (long)gridDim.x * blockDim.x;
  for (; i < n; i += stride) out[i] = in[i];
}

// ---------------------------------------------------------------------------
// F[e*4+b][d] = sum_j w_j * x[b, node_j, d] / sum_j w_j
// ---------------------------------------------------------------------------
__global__ void gather_kernel(const float* __restrict__ x, float* __restrict__ F) {
  int e = blockIdx.x >> 2, b = blockIdx.x & 3;
  int t = threadIdx.x;                       // 256 threads * 4 elems = 1024
  int nn = c_nn[e];
  float winv = c_winv[e];
  const float* xb = x + (size_t)b * Sq * Dm;
  float acc[4] = {0.f, 0.f, 0.f, 0.f};
  for (int j = 0; j < nn; ++j) {
    const float* src = xb + (size_t)c_nodes[e][j] * Dm + t * 4;
    float w = c_w[e][j];
    for (int i = 0; i < 4; ++i) acc[i] += w * src[i];
  }
  float* dst = F + (size_t)blockIdx.x * Dm + t * 4;
  for (int i = 0; i < 4; ++i) dst[i] = acc[i] * winv;
}

// Xq[pi*4+b][d] = x[b, pos[pi], d]
__global__ void pack_q_kernel(const float* __restrict__ x, float* __restrict__ Xq) {
  int pi = blockIdx.x >> 2, b = blockIdx.x & 3;
  int t = threadIdx.x;
  const float* src = x + ((size_t)b * Sq + c_pos[pi]) * Dm + t * 4;
  float* dst = Xq + (size_t)blockIdx.x * Dm + t * 4;
  for (int i = 0; i < 4; ++i) dst[i] = src[i];
}

// ---------------------------------------------------------------------------
// Generic fp32 WMMA GEMM:  Out[m,n] = sum_k A[m,k]*W[n,k] + bias[n]
// One wave per 16x16 output tile; K stepped by 4 via V_WMMA_F32_16X16X4_F32.
// A frag (16x4 f32, 2 VGPR): lane<16 -> K={k,k+1}; lane>=16 -> K={k+2,k+3}.
// B frag mirrors A; C/D: VGPR r holds row r (lanes 0-15) / r+8 (lanes 16-31).
// A-loads are UNGUARDED (see file header): rows >= M feed only discarded D
// rows, and all tile overreads stay inside the workspace allocation.
// ---------------------------------------------------------------------------
__global__ void gemm16_wmma(const float* __restrict__ A, const float* __restrict__ W,
                            const float* __restrict__ bias, float* __restrict__ Out,
                            int M, int N, int K) {
  const int lane = threadIdx.x;
  const int half = lane >> 4;        // 0 or 1
  const int lm   = lane & 15;
  const int n0 = blockIdx.x << 4;
  const int m0 = blockIdx.y << 4;
  const float* Ap = A + (size_t)(m0 + lm) * K + 2 * half;
  const float* Wp = W + (size_t)(n0 + lm) * K + 2 * half;
  v8f acc = {};
#pragma unroll 8
  for (int k = 0; k < K; k += 4) {
    v2f a   = *(const v2f*)(Ap + k);
    v2f bfr = *(const v2f*)(Wp + k);
    acc = __builtin_amdgcn_wmma_f32_16x16x4_f32(
        /*neg_a=*/false, a, /*neg_b=*/false, bfr,
        /*c_mod=*/(short)0, acc, /*reuse_a=*/false, /*reuse_b=*/false);
  }
  const float bv = bias[n0 + lm];
#pragma unroll
  for (int r = 0; r < 8; ++r) {
    int m = m0 + (half << 3) + r;
    if (m < M) Out[(size_t)m * N + (n0 + lm)] = acc[r] + bv;
  }
}

// ---------------------------------------------------------------------------
// Per-row LayerNorm + exact GELU for the lattice MLP (level by row range).
// ---------------------------------------------------------------------------
__global__ void ln_gelu_kernel(const float* __restrict__ In, float* __restrict__ Out,
                               const float* __restrict__ g, const float* __restrict__ bt) {
  int row = blockIdx.x;
  int lvl = row < 28 ? 0 : (row < 52 ? 1 : 2);
  const float* gg = g + lvl * Dm;
  const float* bb = bt + lvl * Dm;
  int t = threadIdx.x;
  float v[4]; float s = 0.f, s2 = 0.f;
  for (int i = 0; i < 4; ++i) {
    float xv = In[(size_t)row * Dm + t * 4 + i];
    v[i] = xv; s += xv; s2 += xv * xv;
  }
  __shared__ float r1[256], r2[256];
  r1[t] = s; r2[t] = s2; __syncthreads();
  for (int o = 128; o > 0; o >>= 1) {
    if (t < o) { r1[t] += r1[t + o]; r2[t] += r2[t + o]; }
    __syncthreads();
  }
  float mu  = r1[0] * (1.f / Dm);
  float var = r2[0] * (1.f / Dm) - mu * mu;
  float inv = rsqrtf(var + 1e-5f);
  for (int i = 0; i < 4; ++i) {
    int d = t * 4 + i;
    float y = (v[i] - mu) * inv * gg[d] + bb[d];
    y = 0.5f * y * (1.f + erff(y * 0.70710678118654752f));   // exact GELU
    Out[(size_t)row * Dm + d] = y;
  }
}

// ---------------------------------------------------------------------------
// Tiny attention: per (pos,b) row; H=4 heads of 256; <=3 keys.
// ---------------------------------------------------------------------------
__global__ void attention_kernel(const float* __restrict__ Qp, const float* __restrict__ Kp,
                                 const float* __restrict__ Vp, float* __restrict__ O) {
  int pi = blockIdx.x >> 2, b = blockIdx.x & 3;
  int row = blockIdx.x;
  int t = threadIdx.x;               // 256 threads * 4 d
  int d0 = t * 4;
  int h = d0 >> 8;                   // head = d/256
  __shared__ float sc[4][4];
  if (t < 16) ((float*)sc)[t] = 0.f;
  __syncthreads();
  int nl = c_nlev[pi];
  float q[4];
  for (int i = 0; i < 4; ++i) q[i] = Qp[(size_t)row * Dm + d0 + i];
  for (int l = 0; l < nl; ++l) {
    int krow = c_kvidx[pi][l] * 4 + b;
    float p = 0.f;
    for (int i = 0; i < 4; ++i) p += q[i] * Kp[(size_t)krow * Dm + d0 + i];
    atomicAdd(&sc[h][l], p);
  }
  __syncthreads();
  float att[3];
  float mx = -1e30f;
  for (int l = 0; l < nl; ++l) { float sv = sc[h][l] * (1.f / 16.f); att[l] = sv; mx = fmaxf(mx, sv); }
  float sum = 0.f;
  for (int l = 0; l < nl; ++l) { att[l] = expf(att[l] - mx); sum += att[l]; }
  float inv = 1.f / sum;
  float o[4] = {0.f, 0.f, 0.f, 0.f};
  for (int l = 0; l < nl; ++l) {
    int krow = c_kvidx[pi][l] * 4 + b;
    float w = att[l] * inv;
    for (int i = 0; i < 4; ++i) o[i] += w * Vp[(size_t)krow * Dm + d0 + i];
  }
  for (int i = 0; i < 4; ++i) O[(size_t)row * Dm + d0 + i] = o[i];
}

// Comb[row][0:1024]=Op[row], [1024:2048]=Xq[row]
__global__ void pack_comb_kernel(const float* __restrict__ Op, const float* __restrict__ Xq,
                                 float* __restrict__ Comb) {
  int row = blockIdx.x, t = threadIdx.x;
  for (int i = 0; i < 8; ++i) {
    int d = t * 8 + i;
    float v = (d < Dm) ? Op[(size_t)row * Dm + d] : Xq[(size_t)row * Dm + (d - Dm)];
    Comb[(size_t)row * (2 * Dm) + d] = v;
  }
}

// Final LN + scatter into the 7 updated rows of out.
__global__ void ln_scatter_kernel(const float* __restrict__ U, const float* __restrict__ g,
                                  const float* __restrict__ bt, float* __restrict__ out) {
  int row = blockIdx.x;
  int pi = row >> 2, b = row & 3;
  int t = threadIdx.x;
  float v[4]; float s = 0.f, s2 = 0.f;
  for (int i = 0; i < 4; ++i) {
    float xv = U[(size_t)row * Dm + t * 4 + i];
    v[i] = xv; s += xv; s2 += xv * xv;
  }
  __shared__ float r1[256], r2[256];
  r1[t] = s; r2[t] = s2; __syncthreads();
  for (int o = 128; o > 0; o >>= 1) {
    if (t < o) { r1[t] += r1[t + o]; r2[t] += r2[t + o]; }
    __syncthreads();
  }
  float mu  = r1[0] * (1.f / Dm);
  float var = r2[0] * (1.f / Dm) - mu * mu;
  float inv = rsqrtf(var + 1e-5f);
  float* dst = out + ((size_t)b * Sq + c_pos[pi]) * Dm;
  for (int i = 0; i < 4; ++i) {
    int d = t * 4 + i;
    dst[d] = (v[i] - mu) * inv * g[d] + bt[d];
  }
}

// ---------------------------------------------------------------------------
// Workspace layout (floats). Padded-tile overreads (up to 4 rows past each
// buffer's valid M) always land in the NEXT buffer, never past OFF_END.
// ---------------------------------------------------------------------------
#define OFF_F     0                       // 64 x 1024
#define OFF_H1    (OFF_F    + 64 * 1024)  // 64 x 1024
#define OFF_H2    (OFF_H1   + 64 * 1024)  // 64 x 1024
#define OFF_KV    (OFF_H2   + 64 * 1024)  // 64 x 1024
#define OFF_XQ    (OFF_KV   + 64 * 1024)  // 28 x 1024
#define OFF_QP    (OFF_XQ   + 28 * 1024)  // 28 x 1024
#define OFF_KP    (OFF_QP   + 28 * 1024)  // 64 x 1024
#define OFF_VP    (OFF_KP   + 64 * 1024)  // 64 x 1024
#define OFF_O     (OFF_VP   + 64 * 1024)  // 28 x 1024
#define OFF_OP    (OFF_O    + 28 * 1024)  // 28 x 1024
#define OFF_COMB  (OFF_OP   + 28 * 1024)  // 28 x 2048
#define OFF_U     (OFF_COMB + 28 * 2048)  // 28 x 1024

extern "C" void kernel_launch(void* const* d_in, const int* in_sizes, int n_in,
                              void* d_out, int out_size, void* d_ws, size_t ws_size,
                              hipStream_t stream) {
  const float* x          = (const float*)d_in[0];
  const float* lt_w1      = (const float*)d_in[1];
  const float* lt_b1      = (const float*)d_in[2];
  const float* lt_ln_g    = (const float*)d_in[3];
  const float* lt_ln_b    = (const float*)d_in[4];
  const float* lt_w2      = (const float*)d_in[5];
  const float* lt_b2      = (const float*)d_in[6];
  const float* attn_in_w  = (const float*)d_in[7];
  const float* attn_in_b  = (const float*)d_in[8];
  const float* attn_out_w = (const float*)d_in[9];
  const float* attn_out_b = (const float*)d_in[10];
  const float* fus_w      = (const float*)d_in[11];
  const float* fus_b      = (const float*)d_in[12];
  const float* fus_ln_g   = (const float*)d_in[13];
  const float* fus_ln_b   = (const float*)d_in[14];
  float* out = (float*)d_out;
  float* ws  = (float*)d_ws;

  float* F    = ws + OFF_F;
  float* H1   = ws + OFF_H1;
  float* H2   = ws + OFF_H2;
  float* KV   = ws + OFF_KV;
  float* Xq   = ws + OFF_XQ;
  float* Qp   = ws + OFF_QP;
  float* Kp   = ws + OFF_KP;
  float* Vp   = ws + OFF_VP;
  float* O    = ws + OFF_O;
  float* Op   = ws + OFF_OP;
  float* Comb = ws + OFF_COMB;
  float* U    = ws + OFF_U;

  // 1) out = x (dominant bandwidth term)
  copy_kernel<<<4096, 256, 0, stream>>>((const float4*)x, (float4*)out,
                                        (long)Bq * Sq * Dm / 4);
  // 2) gather weighted means + pack q rows
  gather_kernel<<<NENT * Bq, 256, 0, stream>>>(x, F);
  pack_q_kernel<<<NPOS * Bq, 256, 0, stream>>>(x, Xq);

  // 3) lattice MLP layer 1 (per-level WMMA GEMMs)  h1 = f @ W1^T + b1
  gemm16_wmma<<<dim3(64, 2), 32, 0, stream>>>(F,            lt_w1,               lt_b1,        H1,            28, Dm, Dm);
  gemm16_wmma<<<dim3(64, 2), 32, 0, stream>>>(F + 28 * Dm,  lt_w1 + 1 * Dm * Dm, lt_b1 + Dm,   H1 + 28 * Dm,  24, Dm, Dm);
  gemm16_wmma<<<dim3(64, 1), 32, 0, stream>>>(F + 52 * Dm,  lt_w1 + 2 * Dm * Dm, lt_b1 + 2*Dm, H1 + 52 * Dm,  12, Dm, Dm);

  // 4) LN + GELU
  ln_gelu_kernel<<<64, 256, 0, stream>>>(H1, H2, lt_ln_g, lt_ln_b);

  // 5) lattice MLP layer 2  kv = h2 @ W2^T + b2
  gemm16_wmma<<<dim3(64, 2), 32, 0, stream>>>(H2,           lt_w2,               lt_b2,        KV,            28, Dm, Dm);
  gemm16_wmma<<<dim3(64, 2), 32, 0, stream>>>(H2 + 28 * Dm, lt_w2 + 1 * Dm * Dm, lt_b2 + Dm,   KV + 28 * Dm,  24, Dm, Dm);
  gemm16_wmma<<<dim3(64, 1), 32, 0, stream>>>(H2 + 52 * Dm, lt_w2 + 2 * Dm * Dm, lt_b2 + 2*Dm, KV + 52 * Dm,  12, Dm, Dm);

  // 6) Q/K/V projections (attn_in_w rows: [0,D)=Wq, [D,2D)=Wk, [2D,3D)=Wv)
  gemm16_wmma<<<dim3(64, 2), 32, 0, stream>>>(Xq, attn_in_w,               attn_in_b,          Qp, 28, Dm, Dm);
  gemm16_wmma<<<dim3(64, 4), 32, 0, stream>>>(KV, attn_in_w + 1 * Dm * Dm, attn_in_b + Dm,     Kp, 64, Dm, Dm);
  gemm16_wmma<<<dim3(64, 4), 32, 0, stream>>>(KV, attn_in_w + 2 * Dm * Dm, attn_in_b + 2 * Dm, Vp, 64, Dm, Dm);

  // 7) tiny softmax attention over <=3 lattice levels
  attention_kernel<<<NPOS * Bq, 256, 0, stream>>>(Qp, Kp, Vp, O);

  // 8) output projection
  gemm16_wmma<<<dim3(64, 2), 32, 0, stream>>>(O, attn_out_w, attn_out_b, Op, 28, Dm, Dm);

  // 9) fuse: [o, q] @ fus_w^T + fus_b  (K = 2048)
  pack_comb_kernel<<<NPOS * Bq, 256, 0, stream>>>(Op, Xq, Comb);
  gemm16_wmma<<<dim3(64, 2), 32, 0, stream>>>(Comb, fus_w, fus_b, U, 28, Dm, 2 * Dm);

  // 10) final LN + scatter into out rows
  ln_scatter_kernel<<<NPOS * Bq, 256, 0, stream>>>(U, fus_ln_g, fus_ln_b, out);
}